// MaxPooling_37546604102082
// MI455X (gfx1250) — compile-verified
//
#include <hip/hip_runtime.h>

// 2x2 max pooling, NCHW fp32.
// Input : (16, 64, 512, 512)  -> 2^28 floats
// Output: (16, 64, 256, 256)  -> 2^26 floats
//
// Pure streaming op: 1 GiB in + 0.25 GiB out, ~0.2 GFLOP. HBM-bound
// (~58 us floor at 23.3 TB/s). Strategy: one float4 of output per thread,
// 4x global_load_b128 + 1x global_store_b128, all non-temporal (no reuse),
// shift/mask-only index math.

typedef float v4f __attribute__((ext_vector_type(4)));

#define IN_W    512u
#define IN_IMG  (512u * 512u)   // floats per (b,c) image
#define OW4     64u             // 256 output cols / 4 per float4
#define OH      256u

__global__ __launch_bounds__(256) void maxpool2x2_nt_kernel(
    const float* __restrict__ x, float* __restrict__ out, unsigned n4)
{
    unsigned t = blockIdx.x * blockDim.x + threadIdx.x;
    if (t >= n4) return;

    // Decode float4-chunk id -> (image, oh, ow4). All powers of two.
    unsigned ow4 = t & (OW4 - 1u);        // chunk within output row
    unsigned tmp = t >> 6;                // /64
    unsigned oh  = tmp & (OH - 1u);       // output row
    unsigned img = tmp >> 8;              // b*c plane index

    const float* p0 = x + (size_t)img * IN_IMG
                        + (size_t)(oh << 1) * IN_W
                        + (size_t)(ow4 << 3);          // 8 input floats per chunk
    const float* p1 = p0 + IN_W;                       // second input row

    // Two B128 loads per row (8 contiguous floats), both rows. 64 B/thread.
    v4f r0a = __builtin_nontemporal_load((const v4f*)p0);
    v4f r0b = __builtin_nontemporal_load(((const v4f*)p0) + 1);
    v4f r1a = __builtin_nontemporal_load((const v4f*)p1);
    v4f r1b = __builtin_nontemporal_load(((const v4f*)p1) + 1);

    // 4-way max per output; linear chain lets backend fuse v_max3_num_f32.
    v4f o;
    o.x = fmaxf(fmaxf(fmaxf(r0a.x, r0a.y), r1a.x), r1a.y);
    o.y = fmaxf(fmaxf(fmaxf(r0a.z, r0a.w), r1a.z), r1a.w);
    o.z = fmaxf(fmaxf(fmaxf(r0b.x, r0b.y), r1b.x), r1b.y);
    o.w = fmaxf(fmaxf(fmaxf(r0b.z, r0b.w), r1b.z), r1b.w);

    __builtin_nontemporal_store(o, ((v4f*)out) + t);
}

extern "C" void kernel_launch(void* const* d_in, const int* in_sizes, int n_in,
                              void* d_out, int out_size, void* d_ws, size_t ws_size,
                              hipStream_t stream) {
    (void)in_sizes; (void)n_in; (void)d_ws; (void)ws_size;
    const float* x = (const float*)d_in[0];
    float* out = (float*)d_out;

    unsigned n4 = (unsigned)(out_size / 4);    // 16,777,216 float4 outputs
    dim3 block(256);                           // 8 wave32s per block
    dim3 grid((n4 + block.x - 1) / block.x);   // 65,536 blocks

    hipLaunchKernelGGL(maxpool2x2_nt_kernel, grid, block, 0, stream, x, out, n4);
}